// KNNModule_41472204210679
// MI455X (gfx1250) — compile-verified
//
#include <hip/hip_runtime.h>

typedef float v2f __attribute__((ext_vector_type(2)));
typedef float v8f __attribute__((ext_vector_type(8)));

#define BATCH   8
#define NPTS    16384
#define NPOINT  2048
#define KNN     32
#define TC      16          // centers per workgroup (one WMMA M-tile)
#define CHUNK   256         // points processed per chunk
#define NCHUNK  (NPTS / CHUNK)
#define DSTRIDE 274         // d2buf row stride in floats (bank-conflict free)
#define THREADS 256

__global__ __launch_bounds__(THREADS)
void knn_wmma_kernel(const float* __restrict__ xyz,
                     const float* __restrict__ center,
                     int* __restrict__ out)
{
    __shared__ float d2buf[2 * TC * DSTRIDE];      // ping-pong distance strips
    __shared__ float mergeD[TC][16][KNN];          // 32 KB
    __shared__ int   mergeI[TC][16][KNN];          // 32 KB
    __shared__ int   heads[TC][16];                // 1 KB

    const int tid  = threadIdx.x;
    const int lane = tid & 31;
    const int wave = tid >> 5;
    const int lo16 = lane & 15;
    const bool hi  = (lane >= 16);

    const int b  = blockIdx.x / (NPOINT / TC);
    const int p0 = (blockIdx.x % (NPOINT / TC)) * TC;

    // ---------------- A operand: 16 centers x (K=4) = (-2cx,-2cy,-2cz, 1) ----
    // ISA 32-bit A 16x4 layout: lanes 0-15 hold K0(V0),K1(V1); lanes 16-31 hold K2,K3.
    const float* cp = center + ((size_t)b * NPOINT + (p0 + lo16)) * 3;
    const float ccx = cp[0], ccy = cp[1], ccz = cp[2];
    v2f amat;
    amat[0] = hi ? (-2.0f * ccz) : (-2.0f * ccx);
    amat[1] = hi ? 1.0f          : (-2.0f * ccy);

    // ---------------- per-thread top-K state (registers, statically indexed) --
    float bd[KNN];
    int   bi[KNN];
#pragma unroll
    for (int k = 0; k < KNN; ++k) { bd[k] = 3.4e38f; bi[k] = -1; }

    const int selm = tid >> 4;   // center row this thread selects for
    const int sels = tid & 15;   // sub-slot within that center (16 threads/center)

    const float* xb = xyz + (size_t)b * NPTS * 3;

    // -------- software pipeline: prefetch chunk 0 points into registers ------
    float ppx[2], ppy[2], ppz[2];
#pragma unroll
    for (int tt = 0; tt < 2; ++tt) {
        const int t = wave * 2 + tt;
        const float* xp = xb + (size_t)(t * 16 + lo16) * 3;
        ppx[tt] = xp[0]; ppy[tt] = xp[1]; ppz[tt] = xp[2];
    }

    for (int ch = 0; ch < NCHUNK; ++ch) {
        const int base = ch * CHUNK;
        float* dbuf = d2buf + (ch & 1) * (TC * DSTRIDE);

        // Build B operands for the current chunk from prefetched registers.
        // B 4x16 layout (mirror of A): lanes0-15 -> K0(V0),K1(V1); lanes16-31 -> K2,K3
        v2f bmat[2];
#pragma unroll
        for (int tt = 0; tt < 2; ++tt) {
            const float px = ppx[tt], py = ppy[tt], pz = ppz[tt];
            const float x2 = px * px + py * py + pz * pz;
            bmat[tt][0] = hi ? pz : px;
            bmat[tt][1] = hi ? x2 : py;
        }

        // Issue next chunk's point loads NOW; latency hidden by WMMA+selection.
        if (ch + 1 < NCHUNK) {
#pragma unroll
            for (int tt = 0; tt < 2; ++tt) {
                const int t = wave * 2 + tt;
                const float* xp = xb + (size_t)((ch + 1) * CHUNK + t * 16 + lo16) * 3;
                ppx[tt] = xp[0]; ppy[tt] = xp[1]; ppz[tt] = xp[2];
            }
        }

        // -------- produce d2 strip: s[m][n] = ||x||^2 - 2 c.x via WMMA -------
#pragma unroll
        for (int tt = 0; tt < 2; ++tt) {
            const int t = wave * 2 + tt;
            v8f acc = {};
            acc = __builtin_amdgcn_wmma_f32_16x16x4_f32(
                      false, amat, false, bmat[tt], (short)0, acc, false, false);

            // D 16x16 layout: elem v -> M = v + (hi?8:0), N = lane&15
            const int colb = t * 16 + lo16;
            const int mo   = hi ? 8 : 0;
#pragma unroll
            for (int v = 0; v < 8; ++v)
                dbuf[(v + mo) * DSTRIDE + colb] = acc[v];
        }

        __syncthreads();   // single barrier per chunk (ping-pong makes it safe)

        // -------- selection: 16 threads per center, 16 candidates each -------
#pragma unroll
        for (int i = 0; i < 16; ++i) {
            const int col = sels + i * 16;
            const float d = dbuf[selm * DSTRIDE + col];
            const int  id = base + col;
            if (d < bd[KNN - 1]) {               // threshold reject is O(1)
                bd[KNN - 1] = d; bi[KNN - 1] = id;
#pragma unroll
                for (int k = KNN - 1; k > 0; --k) {   // one bubble pass inserts
                    if (bd[k] < bd[k - 1]) {
                        const float td = bd[k]; bd[k] = bd[k - 1]; bd[k - 1] = td;
                        const int   ti = bi[k]; bi[k] = bi[k - 1]; bi[k - 1] = ti;
                    }
                }
            }
        }
    }

    // ---------------- final exact merge: 16 sorted lists of 32 per center ----
    __syncthreads();       // last selection done on all threads before merge dump
#pragma unroll
    for (int k = 0; k < KNN; ++k) {
        mergeD[selm][sels][k] = bd[k];
        mergeI[selm][sels][k] = bi[k];
    }
    if (sels == 0) {
#pragma unroll
        for (int j = 0; j < 16; ++j) heads[selm][j] = 0;
    }
    __syncthreads();

    if (tid < TC) {
        const int m = tid;
        int* op = out + (((size_t)b * NPOINT) + p0 + m) * KNN;
        for (int k = 0; k < KNN; ++k) {
            float bestd = 3.5e38f;
            int   bestj = 0;
            for (int j = 0; j < 16; ++j) {
                const int h = heads[m][j];
                const float d = (h < KNN) ? mergeD[m][j][h] : 3.5e38f;
                if (d < bestd) { bestd = d; bestj = j; }
            }
            const int h = heads[m][bestj];
            op[k] = mergeI[m][bestj][h];
            heads[m][bestj] = h + 1;
        }
    }
}

extern "C" void kernel_launch(void* const* d_in, const int* in_sizes, int n_in,
                              void* d_out, int out_size, void* d_ws, size_t ws_size,
                              hipStream_t stream) {
    const float* xyz    = (const float*)d_in[0];   // [8,16384,3] f32
    const float* center = (const float*)d_in[1];   // [8,2048,3]  f32
    int* out = (int*)d_out;                        // [8,2048,32] int32 indices
    (void)in_sizes; (void)n_in; (void)out_size; (void)d_ws; (void)ws_size;

    dim3 grid(BATCH * (NPOINT / TC));              // 1024 workgroups
    knn_wmma_kernel<<<grid, THREADS, 0, stream>>>(xyz, center, out);
}